// FeatureQuantizationV2_90366111908646
// MI455X (gfx1250) — compile-verified
//
#include <hip/hip_runtime.h>
#include <hip/hip_bf16.h>
#include <stdint.h>

#define QTHREADS 128  // F/4 lanes, one 16B vector per lane (F == 512)

typedef float v4f __attribute__((ext_vector_type(4)));  // clang vector: valid for
                                                        // __builtin_nontemporal_*

// --- clear degree + presence scratch (must be zeroed every call) ---
__global__ void zero_ws_kernel(unsigned* __restrict__ ws, int n) {
    int i = blockIdx.x * blockDim.x + threadIdx.x;
    if (i < n) ws[i] = 0u;
}

// --- in-degree histogram: deg[dst[e]] += 1 (L2-resident atomics) ---
__global__ void degree_kernel(const int* __restrict__ dst,
                              unsigned* __restrict__ deg, int n_edges) {
    int i = blockIdx.x * blockDim.x + threadIdx.x;
    if (i < n_edges) {
        int d = __builtin_nontemporal_load(&dst[i]);   // edge list read exactly once
        atomicAdd(&deg[d], 1u);
    }
}

// --- per-node STE quantization, one block per node ---
__global__ __launch_bounds__(QTHREADS)
void quantize_kernel(const float* __restrict__ fea,
                     const float* __restrict__ gama,
                     const float* __restrict__ bitw,
                     const unsigned* __restrict__ deg,
                     unsigned* __restrict__ present,
                     float* __restrict__ out,
                     int F, int K) {
    const int node = blockIdx.x;
    const int tid  = threadIdx.x;

    __shared__ v4f   tile[QTHREADS];    // one 2KB feature row
    __shared__ float params[4];         // s, rcp(s), qmin, qmax (uniform per node)

    // CDNA5 async global->LDS copy: each of the 128 lanes moves 16 bytes,
    // non-temporal (fea is streamed exactly once; don't churn L2).
    // ASYNCcnt tracks completion; no VGPR round-trip on the load side.
    const float* gsrc   = fea + (size_t)node * F + tid * 4;
    unsigned     ldsoff = (unsigned)(size_t)(&tile[tid]);  // low 32 bits = LDS offset
    asm volatile("global_load_async_to_lds_b128 %0, %1, off th:TH_LOAD_NT"
                 :: "v"(ldsoff), "v"(gsrc) : "memory");

    // Overlap: lane 0 resolves per-node scale/bit while the DMA is in flight.
    // One correctly-rounded divide per ROW (not per element) keeps the inner
    // loop at ~5 VALU ops/element so the kernel stays HBM-bound.
    if (tid == 0) {
        int si = (int)deg[node];
        si = si > K - 1 ? K - 1 : si;         // deg >= 0 always
        float s = fabsf(gama[si]);            // (K,1) table, flat
        float b = rintf(bitw[si]);            // STE round (fwd = round-to-even)
        float p = exp2f(b - 1.0f);            // 2^(b-1), exact for integer b
        params[0] = s;
        params[1] = 1.0f / s;                 // IEEE-rounded reciprocal, once per row
        params[2] = -p;                       // qmin
        params[3] = p - 1.0f;                 // qmax
        present[si] = 1u;                     // presence mask (benign race: all write 1)
    }

    asm volatile("s_wait_asynccnt 0" ::: "memory");
    __syncthreads();

    const float s    = params[0];
    const float rcp  = params[1];
    const float qmin = params[2];
    const float qmax = params[3];

    v4f x = tile[tid];                        // ds_load_b128
    v4f r;
    r.x = rintf(fminf(fmaxf(x.x * rcp, qmin), qmax)) * s;
    r.y = rintf(fminf(fmaxf(x.y * rcp, qmin), qmax)) * s;
    r.z = rintf(fminf(fmaxf(x.z * rcp, qmin), qmax)) * s;
    r.w = rintf(fminf(fmaxf(x.w * rcp, qmin), qmax)) * s;

    // Output is written once and never re-read: non-temporal b128 store.
    __builtin_nontemporal_store(r, (v4f*)(out + (size_t)node * F + tid * 4));
}

// --- bit_sum = F * sum(bit[i] * present[i]) / 8 / 1024 ---
__global__ void bitsum_kernel(const float* __restrict__ bitw,
                              const unsigned* __restrict__ present,
                              float* __restrict__ out_scalar, int F, int K) {
    __shared__ float red[256];
    int t = threadIdx.x;
    float acc = 0.0f;
    for (int i = t; i < K; i += blockDim.x)
        acc += present[i] ? bitw[i] : 0.0f;   // exact: multiples of 4.0
    red[t] = acc;
    __syncthreads();
    for (int sdx = blockDim.x >> 1; sdx > 0; sdx >>= 1) {
        if (t < sdx) red[t] += red[t + sdx];
        __syncthreads();
    }
    if (t == 0) *out_scalar = (float)F * red[0] / 8.0f / 1024.0f;
}

extern "C" void kernel_launch(void* const* d_in, const int* in_sizes, int n_in,
                              void* d_out, int out_size, void* d_ws, size_t ws_size,
                              hipStream_t stream) {
    const float* fea  = (const float*)d_in[0];   // (N, F) float32
    const int*   ei   = (const int*)d_in[1];     // (2, E) int32, row-major
    const float* gama = (const float*)d_in[2];   // (K, 1) float32
    const float* bitw = (const float*)d_in[3];   // (K, 1) float32
    float*       out  = (float*)d_out;           // [N*F fea_q] ++ [1 bit_sum]

    const int K = in_sizes[2];                   // 512 degree bins
    const int F = K;                             // N_FEAT == input_dim
    const int N = in_sizes[0] / F;               // 100000 nodes
    const int E = in_sizes[1] / 2;               // 3200000 edges

    unsigned* deg     = (unsigned*)d_ws;         // N u32
    unsigned* present = deg + N;                 // K u32

    const int zn = N + K;
    zero_ws_kernel<<<(zn + 255) / 256, 256, 0, stream>>>((unsigned*)d_ws, zn);
    degree_kernel<<<(E + 255) / 256, 256, 0, stream>>>(ei + E /* row 1 */, deg, E);
    quantize_kernel<<<N, F / 4, 0, stream>>>(fea, gama, bitw, deg, present, out, F, K);
    bitsum_kernel<<<1, 256, 0, stream>>>(bitw, present, out + (size_t)N * F, F, K);
}